// decoder_18064632447253
// MI455X (gfx1250) — compile-verified
//
#include <hip/hip_runtime.h>
#include <hip/hip_bf16.h>

// Problem constants (from reference)
#define T_LEN 1000
#define B_SZ  16
#define D_SZ  512
#define S_LEN 100
#define V_SZ  4004
#define SOS   4001
#define TB    (T_LEN * B_SZ)

typedef float v2f __attribute__((ext_vector_type(2)));
typedef float v8f __attribute__((ext_vector_type(8)));

// ---------------------------------------------------------------------------
// fp32 WMMA 16x16x4 helpers (V_WMMA_F32_16X16X4_F32, wave32)
// A (16x4, MxK) fragment: lanes 0-15 -> row=lane, K={k0,k0+1};
//                         lanes 16-31 -> row=lane-16, K={k0+2,k0+3}.
// B (4x16, KxN) fragment for B[k][n] = W[n][k] (row-major W, "X @ W.T"):
//   identical load pattern with row interpreted as N.
// C/D (16x16): lane<16 -> col=lane, rows 0..7 in v[0..7];
//              lane>=16 -> col=lane-16, rows 8..15.
// ---------------------------------------------------------------------------
__device__ __forceinline__ v2f ld_frag(const float* __restrict__ base, int ld, int lane) {
  int row = lane & 15;
  int kk  = (lane >> 4) << 1;           // 0 for lanes 0-15, 2 for lanes 16-31
  const float* p = base + (size_t)row * ld + kk;
  v2f r; r.x = p[0]; r.y = p[1];
  return r;
}

__device__ __forceinline__ v8f wmma4(v2f a, v2f b, v8f c) {
  return __builtin_amdgcn_wmma_f32_16x16x4_f32(false, a, false, b, (short)0, c,
                                               false, false);
}

__device__ __forceinline__ float sigmoidf_(float x) {
  return 1.0f / (1.0f + __expf(-x));
}

// ---------------------------------------------------------------------------
// Init: h0 = 0, c = 0, alpha = 0   (once per launch; ws is poisoned)
// ---------------------------------------------------------------------------
__global__ void k_init(float* __restrict__ h0, float* __restrict__ c,
                       float* __restrict__ alpha) {
  int i = blockIdx.x * blockDim.x + threadIdx.x;
  if (i < B_SZ * D_SZ) { h0[i] = 0.0f; c[i] = 0.0f; }
  if (i < TB)          { alpha[i] = 0.0f; }
}

// ---------------------------------------------------------------------------
// phi[t*B+b, e] = sum_d H[t*B+b, d] * phi_w[e, d] + phi_b[e]
// grid = (TB/16, D/64), 1 wave/block, 4 N-tiles per wave (A reuse)
// ---------------------------------------------------------------------------
__global__ __launch_bounds__(32) void k_phi(const float* __restrict__ H,
                                            const float* __restrict__ Wphi,
                                            const float* __restrict__ bphi,
                                            float* __restrict__ phi) {
  int lane = threadIdx.x;
  int r0 = blockIdx.x * 16;
  int n0 = blockIdx.y * 64;
  v8f acc[4] = {};
  for (int k0 = 0; k0 < D_SZ; k0 += 4) {
    v2f a = ld_frag(H + (size_t)r0 * D_SZ + k0, D_SZ, lane);
#pragma unroll
    for (int j = 0; j < 4; ++j) {
      v2f b = ld_frag(Wphi + (size_t)(n0 + 16 * j) * D_SZ + k0, D_SZ, lane);
      acc[j] = wmma4(a, b, acc[j]);
    }
  }
  int col  = lane & 15;
  int row0 = (lane >> 4) << 3;
#pragma unroll
  for (int j = 0; j < 4; ++j) {
    int cg = n0 + 16 * j + col;
    float bv = bphi[cg];
#pragma unroll
    for (int i = 0; i < 8; ++i)
      phi[(size_t)(r0 + row0 + i) * D_SZ + cg] = acc[j][i] + bv;
  }
}

// ---------------------------------------------------------------------------
// Fused LSTM cell. One wave per 16-column slice of D; computes the i/f/g/o
// tiles for that slice (8 WMMAs per K-step: x-path + h-path for 4 gates),
// then applies the nonlinearity and updates c (in place) and h_out.
// yi is gathered from emb via the teacher-forcing token (no materialization).
// ---------------------------------------------------------------------------
__global__ __launch_bounds__(32) void k_lstm(const float* __restrict__ emb,
                                             const int*   __restrict__ wt,
                                             const float* __restrict__ w_ih,
                                             const float* __restrict__ w_hh,
                                             const float* __restrict__ b_ih,
                                             const float* __restrict__ b_hh,
                                             const float* __restrict__ h_in,
                                             float* __restrict__ h_out,
                                             float* __restrict__ c, int s) {
  int lane = threadIdx.x;
  int n0 = blockIdx.x * 16;
  int myrow = lane & 15;                       // batch row for A fragments
  int kk = (lane >> 4) << 1;
  int tok = (s == 0) ? SOS : wt[myrow * S_LEN + (s - 1)];
  const float* xrow = emb + (size_t)tok * D_SZ;
  const float* hrow = h_in + (size_t)myrow * D_SZ;

  v8f acc[4] = {};
  for (int k0 = 0; k0 < D_SZ; k0 += 4) {
    v2f ax; ax.x = xrow[k0 + kk]; ax.y = xrow[k0 + kk + 1];
    v2f ah; ah.x = hrow[k0 + kk]; ah.y = hrow[k0 + kk + 1];
#pragma unroll
    for (int g = 0; g < 4; ++g) {
      v2f bx = ld_frag(w_ih + (size_t)(g * D_SZ + n0) * D_SZ + k0, D_SZ, lane);
      acc[g] = wmma4(ax, bx, acc[g]);
      v2f bh = ld_frag(w_hh + (size_t)(g * D_SZ + n0) * D_SZ + k0, D_SZ, lane);
      acc[g] = wmma4(ah, bh, acc[g]);
    }
  }
  int col  = lane & 15;
  int row0 = (lane >> 4) << 3;
  int cg = n0 + col;
  float bi = b_ih[cg]            + b_hh[cg];
  float bf = b_ih[D_SZ + cg]     + b_hh[D_SZ + cg];
  float bg = b_ih[2 * D_SZ + cg] + b_hh[2 * D_SZ + cg];
  float bo = b_ih[3 * D_SZ + cg] + b_hh[3 * D_SZ + cg];
#pragma unroll
  for (int i = 0; i < 8; ++i) {
    int bb = row0 + i;
    float gi = sigmoidf_(acc[0][i] + bi);
    float gf = sigmoidf_(acc[1][i] + bf);
    float gg = tanhf(acc[2][i] + bg);
    float go = sigmoidf_(acc[3][i] + bo);
    size_t idx = (size_t)bb * D_SZ + cg;
    float cn = gf * c[idx] + gi * gg;
    c[idx] = cn;
    h_out[idx] = go * tanhf(cn);
  }
}

// ---------------------------------------------------------------------------
// psi[b,e] = sum_d h[b,d] * psi_w[e,d] + psi_b[e]   (grid = 8, 4 tiles/wave)
// ---------------------------------------------------------------------------
__global__ __launch_bounds__(32) void k_psi(const float* __restrict__ h,
                                            const float* __restrict__ Wpsi,
                                            const float* __restrict__ bpsi,
                                            float* __restrict__ psi) {
  int lane = threadIdx.x;
  int n0 = blockIdx.x * 64;
  v8f acc[4] = {};
  for (int k0 = 0; k0 < D_SZ; k0 += 4) {
    v2f a = ld_frag(h + k0, D_SZ, lane);
#pragma unroll
    for (int j = 0; j < 4; ++j) {
      v2f b = ld_frag(Wpsi + (size_t)(n0 + 16 * j) * D_SZ + k0, D_SZ, lane);
      acc[j] = wmma4(a, b, acc[j]);
    }
  }
  int col  = lane & 15;
  int row0 = (lane >> 4) << 3;
#pragma unroll
  for (int j = 0; j < 4; ++j) {
    int cg = n0 + 16 * j + col;
    float bv = bpsi[cg];
#pragma unroll
    for (int i = 0; i < 8; ++i)
      psi[(size_t)(row0 + i) * D_SZ + cg] = acc[j][i] + bv;
  }
}

// ---------------------------------------------------------------------------
// e[b,t] = attn_b + sum_d attn_w[d] * tanh(phi[t,b,d] + psi[b,d] + conv(d,t,b))
// conv(d,t,b) = conv_b[d] + sum_{k=0..10} conv_w[d,k] * alpha[t-5+k, b]
// One wave per (t,b); 16 d-elements per lane; wave shuffle reduction.
// ---------------------------------------------------------------------------
__global__ __launch_bounds__(256) void k_e(const float* __restrict__ phi,
                                           const float* __restrict__ psi,
                                           const float* __restrict__ alpha,
                                           const float* __restrict__ conv_w,
                                           const float* __restrict__ conv_b,
                                           const float* __restrict__ attn_w,
                                           const float* __restrict__ attn_b,
                                           float* __restrict__ e) {
  int lane = threadIdx.x & 31;
  int wave = (blockIdx.x * blockDim.x + threadIdx.x) >> 5;
  if (wave >= TB) return;
  int t = wave / B_SZ, b = wave % B_SZ;

  float win[11];
#pragma unroll
  for (int k = 0; k < 11; ++k) {
    int t2 = t - 5 + k;
    win[k] = (t2 >= 0 && t2 < T_LEN) ? alpha[(size_t)t2 * B_SZ + b] : 0.0f;
  }
  const float* ph = phi + (size_t)wave * D_SZ;
  const float* ps = psi + (size_t)b * D_SZ;

  float sum = 0.0f;
#pragma unroll 4
  for (int j = 0; j < 16; ++j) {
    int d = lane + 32 * j;
    const float* cw = conv_w + (size_t)d * 11;
    float cv = conv_b[d];
#pragma unroll
    for (int k = 0; k < 11; ++k) cv += cw[k] * win[k];
    sum += attn_w[d] * tanhf(ph[d] + ps[d] + cv);
  }
#pragma unroll
  for (int off = 16; off; off >>= 1) sum += __shfl_xor(sum, off, 32);
  if (lane == 0) e[(size_t)b * T_LEN + t] = sum + attn_b[0];
}

// ---------------------------------------------------------------------------
// Per-b: softmax over T (writes normalized alpha[t,b]) and
// ci[b,d] = sum_t alpha[t,b] * H[t,b,d].  16 blocks x 256 threads.
// ---------------------------------------------------------------------------
__global__ __launch_bounds__(256) void k_attn(const float* __restrict__ e,
                                              const float* __restrict__ H,
                                              float* __restrict__ alpha,
                                              float* __restrict__ ci) {
  int b = blockIdx.x, tid = threadIdx.x;
  __shared__ float red[256];
  const float* eb = e + (size_t)b * T_LEN;

  float m = -1e30f;
  for (int t = tid; t < T_LEN; t += 256) m = fmaxf(m, eb[t]);
  red[tid] = m; __syncthreads();
  for (int st = 128; st; st >>= 1) {
    if (tid < st) red[tid] = fmaxf(red[tid], red[tid + st]);
    __syncthreads();
  }
  m = red[0]; __syncthreads();

  float ssum = 0.0f;
  for (int t = tid; t < T_LEN; t += 256) {
    float v = __expf(eb[t] - m);
    alpha[(size_t)t * B_SZ + b] = v;     // unnormalized for now
    ssum += v;
  }
  red[tid] = ssum; __syncthreads();
  for (int st = 128; st; st >>= 1) {
    if (tid < st) red[tid] += red[tid + st];
    __syncthreads();
  }
  float inv = 1.0f / red[0];
  __syncthreads();

  for (int t = tid; t < T_LEN; t += 256) alpha[(size_t)t * B_SZ + b] *= inv;
  __syncthreads();

  // context: threads own d = tid and tid+256
  float a0 = 0.0f, a1 = 0.0f;
  for (int t = 0; t < T_LEN; ++t) {
    float av = alpha[(size_t)t * B_SZ + b];           // broadcast read
    const float* Hr = H + (size_t)(t * B_SZ + b) * D_SZ;
    __builtin_prefetch(Hr + B_SZ * D_SZ, 0, 0);       // next t row
    a0 += av * Hr[tid];
    a1 += av * Hr[tid + 256];
  }
  ci[(size_t)b * D_SZ + tid]       = a0;
  ci[(size_t)b * D_SZ + tid + 256] = a1;
}

// ---------------------------------------------------------------------------
// logits[b,v] = wdist_b[v] + sum_d ci[b,d]*Wd[v,d] + sum_d h[b,d]*Wd[v,512+d]
// grid = 63 blocks x 4 tiles (covers 4032 >= 4004, tail masked/clamped)
// ---------------------------------------------------------------------------
__global__ __launch_bounds__(32) void k_logits(const float* __restrict__ ci,
                                               const float* __restrict__ h,
                                               const float* __restrict__ Wd,
                                               const float* __restrict__ bd,
                                               float* __restrict__ logits) {
  int lane = threadIdx.x;
  int n0 = blockIdx.x * 64;
  int myrow = lane & 15;
  int kk = (lane >> 4) << 1;
  v8f acc[4] = {};
  for (int k0 = 0; k0 < D_SZ; k0 += 4) {
    v2f aci; { const float* p = ci + (size_t)myrow * D_SZ + k0 + kk; aci.x = p[0]; aci.y = p[1]; }
    v2f ahh; { const float* p = h  + (size_t)myrow * D_SZ + k0 + kk; ahh.x = p[0]; ahh.y = p[1]; }
#pragma unroll
    for (int j = 0; j < 4; ++j) {
      int nr = n0 + 16 * j + (lane & 15);
      if (nr >= V_SZ) nr = V_SZ - 1;                 // clamp: stays in-bounds
      const float* w = Wd + (size_t)nr * (2 * D_SZ);
      v2f b1; b1.x = w[k0 + kk];        b1.y = w[k0 + kk + 1];
      acc[j] = wmma4(aci, b1, acc[j]);
      v2f b2; b2.x = w[D_SZ + k0 + kk]; b2.y = w[D_SZ + k0 + kk + 1];
      acc[j] = wmma4(ahh, b2, acc[j]);
    }
  }
  int col  = lane & 15;
  int row0 = (lane >> 4) << 3;
#pragma unroll
  for (int j = 0; j < 4; ++j) {
    int cg = n0 + 16 * j + col;
    if (cg >= V_SZ) continue;                        // mask tail columns
    float bv = bd[cg];
#pragma unroll
    for (int i = 0; i < 8; ++i)
      logits[(size_t)(row0 + i) * V_SZ + cg] = acc[j][i] + bv;
  }
}

// ---------------------------------------------------------------------------
// probs = softmax over V per b, written straight to d_out (step offset)
// ---------------------------------------------------------------------------
__global__ __launch_bounds__(256) void k_probs(const float* __restrict__ logits,
                                               float* __restrict__ out) {
  int b = blockIdx.x, tid = threadIdx.x;
  __shared__ float red[256];
  const float* lb = logits + (size_t)b * V_SZ;
  float* ob = out + (size_t)b * V_SZ;

  float m = -1e30f;
  for (int v = tid; v < V_SZ; v += 256) m = fmaxf(m, lb[v]);
  red[tid] = m; __syncthreads();
  for (int st = 128; st; st >>= 1) {
    if (tid < st) red[tid] = fmaxf(red[tid], red[tid + st]);
    __syncthreads();
  }
  m = red[0]; __syncthreads();

  float ssum = 0.0f;
  for (int v = tid; v < V_SZ; v += 256) {
    float ex = __expf(lb[v] - m);
    ob[v] = ex;
    ssum += ex;
  }
  red[tid] = ssum; __syncthreads();
  for (int st = 128; st; st >>= 1) {
    if (tid < st) red[tid] += red[tid + st];
    __syncthreads();
  }
  float inv = 1.0f / red[0];
  __syncthreads();
  for (int v = tid; v < V_SZ; v += 256) ob[v] *= inv;
}

// ---------------------------------------------------------------------------
// Host: 1 init + 1 phi + 6 kernels x 100 steps, all on `stream`
// ---------------------------------------------------------------------------
extern "C" void kernel_launch(void* const* d_in, const int* in_sizes, int n_in,
                              void* d_out, int out_size, void* d_ws, size_t ws_size,
                              hipStream_t stream) {
  const float* H      = (const float*)d_in[0];
  const int*   wt     = (const int*)  d_in[1];   // Word_target [B,S]
  const float* conv_w = (const float*)d_in[2];
  const float* conv_b = (const float*)d_in[3];
  const float* psi_w  = (const float*)d_in[4];
  const float* psi_b  = (const float*)d_in[5];
  const float* phi_w  = (const float*)d_in[6];
  const float* phi_b  = (const float*)d_in[7];
  const float* attn_w = (const float*)d_in[8];
  const float* attn_b = (const float*)d_in[9];
  const float* emb    = (const float*)d_in[10];
  const float* w_ih   = (const float*)d_in[11];
  const float* w_hh   = (const float*)d_in[12];
  const float* b_ih   = (const float*)d_in[13];
  const float* b_hh   = (const float*)d_in[14];
  const float* wdist_w= (const float*)d_in[15];
  const float* wdist_b= (const float*)d_in[16];
  float* out = (float*)d_out;

  // Workspace layout (floats)
  float* ws = (float*)d_ws;
  float* phi    = ws;                     size_t off = (size_t)TB * D_SZ;
  float* h0     = ws + off;               off += B_SZ * D_SZ;
  float* h1     = ws + off;               off += B_SZ * D_SZ;
  float* c      = ws + off;               off += B_SZ * D_SZ;
  float* psi    = ws + off;               off += B_SZ * D_SZ;
  float* ci     = ws + off;               off += B_SZ * D_SZ;
  float* alpha  = ws + off;               off += TB;
  float* e      = ws + off;               off += (size_t)B_SZ * T_LEN;
  float* logits = ws + off;               off += (size_t)B_SZ * V_SZ;
  float* hbuf[2] = { h0, h1 };

  // state init (ws is poisoned before timing)
  k_init<<<(TB + 255) / 256, 256, 0, stream>>>(h0, c, alpha);

  // loop-invariant phi = H @ phi_w.T + phi_b
  k_phi<<<dim3(TB / 16, D_SZ / 64), 32, 0, stream>>>(H, phi_w, phi_b, phi);

  for (int s = 0; s < S_LEN; ++s) {
    const float* h_in = hbuf[s & 1];
    float* h_cur = hbuf[(s + 1) & 1];

    k_lstm<<<D_SZ / 16, 32, 0, stream>>>(emb, wt, w_ih, w_hh, b_ih, b_hh,
                                         h_in, h_cur, c, s);
    k_psi<<<D_SZ / 64, 32, 0, stream>>>(h_cur, psi_w, psi_b, psi);
    k_e<<<TB / 8, 256, 0, stream>>>(phi, psi, alpha, conv_w, conv_b,
                                    attn_w, attn_b, e);
    k_attn<<<B_SZ, 256, 0, stream>>>(e, H, alpha, ci);
    k_logits<<<63, 32, 0, stream>>>(ci, h_cur, wdist_w, wdist_b, logits);
    k_probs<<<B_SZ, 256, 0, stream>>>(logits, out + (size_t)s * B_SZ * V_SZ);
  }
}